// LowRankConv2d_89352499626592
// MI455X (gfx1250) — compile-verified
//
#include <hip/hip_runtime.h>

typedef __attribute__((ext_vector_type(2))) float v2f;
typedef __attribute__((ext_vector_type(8))) float v8f;

#define HWSZ  3136   // 56*56
#define WID   56
#define CIN   256
#define COUT  256
#define RANK  32
#define KTOT  288    // 9*32
#define NTOT  50176  // 16*3136

// ---------------------------------------------------------------------------
// Stage 1: U[(k*32+r)][n] = sum_c G[k][r][c] * x[b][c][p],  n = b*3136 + p
// GEMM: M=288 (k,r), K=256 (c), N=50176 (b,p).
// Block = 128 threads (4 waves). Block computes 32(M) x 64(N).
// Each wave: 32(M) x 16(N) via two accumulators sharing B loads.
// ---------------------------------------------------------------------------
__global__ __launch_bounds__(128) void lr_stage1(const float* __restrict__ x,
                                                 const float* __restrict__ G,
                                                 float* __restrict__ U)
{
    __shared__ float Gs[32 * 257];   // 32 A-rows, padded stride (bank-conflict free)
    const int tid = threadIdx.x;
    const int m0  = blockIdx.x * 32;            // 288/32 = 9 blocks in M

    // cooperative, coalesced load of 32x256 A tile
    for (int i = tid; i < 32 * 256; i += 128) {
        int row = i >> 8, col = i & 255;
        Gs[row * 257 + col] = G[(m0 + row) * 256 + col];
    }
    __syncthreads();

    const int wave = tid >> 5;
    const int lane = tid & 31;
    const int col  = lane & 15;                 // N index within tile / A row (lanes mod 16)
    const int klo  = (lane >> 4) << 1;          // 0 for lanes 0-15, 2 for lanes 16-31

    const int n0   = (blockIdx.y * 4 + wave) * 16;
    const int bidx = n0 / HWSZ;                 // 16-wide tile never straddles batch
    const int p0   = n0 - bidx * HWSZ;
    const float* xb = x + (size_t)bidx * CIN * HWSZ + p0 + col;

    v8f acc0 = {};
    v8f acc1 = {};
    #pragma unroll 8
    for (int k0 = 0; k0 < CIN; k0 += 4) {
        v2f b;
        b.x = xb[(size_t)(k0 + klo)     * HWSZ];
        b.y = xb[(size_t)(k0 + klo + 1) * HWSZ];
        v2f a0, a1;
        a0.x = Gs[col * 257 + k0 + klo];
        a0.y = Gs[col * 257 + k0 + klo + 1];
        a1.x = Gs[(col + 16) * 257 + k0 + klo];
        a1.y = Gs[(col + 16) * 257 + k0 + klo + 1];
        acc0 = __builtin_amdgcn_wmma_f32_16x16x4_f32(false, a0, false, b, (short)0, acc0, false, false);
        acc1 = __builtin_amdgcn_wmma_f32_16x16x4_f32(false, a1, false, b, (short)0, acc1, false, false);
    }

    const int mo = (lane >> 4) << 3;            // +8 for upper half-wave
    float* U0 = U + (size_t)(m0 + mo) * NTOT + n0 + col;
    float* U1 = U0 + (size_t)16 * NTOT;
    #pragma unroll
    for (int v = 0; v < 8; ++v) {
        U0[(size_t)v * NTOT] = acc0[v];
        U1[(size_t)v * NTOT] = acc1[v];
    }
}

// ---------------------------------------------------------------------------
// Stage 2: out[b][o][p] = (1/9) * sum_{k,r} Hm[k][o][r] * U[k*32+r][shift_k(n)] + bias[o]
// GEMM: M=256 (o), K=288 (k,r), N=50176, with per-k shifted/halo-masked B gather.
// Block = 128 threads (4 waves), 32(M) x 64(N) per block.
// ---------------------------------------------------------------------------
__global__ __launch_bounds__(128) void lr_stage2(const float* __restrict__ U,
                                                 const float* __restrict__ Hm,
                                                 const float* __restrict__ bias,
                                                 float* __restrict__ out)
{
    __shared__ float Hs[32 * 289];   // 32 A-rows (o), K = 288, padded stride
    const int tid = threadIdx.x;
    const int m0  = blockIdx.x * 32;            // 256/32 = 8 blocks in M

    for (int i = tid; i < 32 * KTOT; i += 128) {
        int o_l = i / KTOT;
        int kk  = i - o_l * KTOT;
        int k = kk >> 5, r = kk & 31;
        Hs[o_l * 289 + kk] = Hm[(size_t)k * (COUT * RANK) + (size_t)(m0 + o_l) * RANK + r];
    }
    __syncthreads();

    const int wave = tid >> 5;
    const int lane = tid & 31;
    const int col  = lane & 15;
    const int klo  = (lane >> 4) << 1;

    const int n0   = (blockIdx.y * 4 + wave) * 16;
    const int n    = n0 + col;
    const int bidx = n / HWSZ;
    const int p    = n - bidx * HWSZ;
    const int h    = p / WID;
    const int w    = p - h * WID;

    // Precompute the 9 gathered column offsets + validity (clamped: no OOB loads)
    int offs[9];
    unsigned vmask = 0;
    #pragma unroll
    for (int k = 0; k < 9; ++k) {
        int ki = k / 3, kj = k - ki * 3;
        int hs = h + ki - 1, ws = w + kj - 1;
        bool ok = (hs >= 0) & (hs < WID) & (ws >= 0) & (ws < WID);
        offs[k] = ok ? (bidx * HWSZ + hs * WID + ws) : 0;
        vmask  |= (ok ? 1u : 0u) << k;
    }

    v8f acc0 = {};
    v8f acc1 = {};
    #pragma unroll
    for (int k = 0; k < 9; ++k) {
        const bool ok = (vmask >> k) & 1u;
        const float* Ucol = U + offs[k];
        #pragma unroll
        for (int r0 = 0; r0 < RANK; r0 += 4) {
            const int kk = k * RANK + r0 + klo;
            v2f b;
            b.x = ok ? Ucol[(size_t)kk       * NTOT] : 0.0f;
            b.y = ok ? Ucol[(size_t)(kk + 1) * NTOT] : 0.0f;
            v2f a0, a1;
            a0.x = Hs[col * 289 + kk];
            a0.y = Hs[col * 289 + kk + 1];
            a1.x = Hs[(col + 16) * 289 + kk];
            a1.y = Hs[(col + 16) * 289 + kk + 1];
            acc0 = __builtin_amdgcn_wmma_f32_16x16x4_f32(false, a0, false, b, (short)0, acc0, false, false);
            acc1 = __builtin_amdgcn_wmma_f32_16x16x4_f32(false, a1, false, b, (short)0, acc1, false, false);
        }
    }

    const int mo = (lane >> 4) << 3;
    const float inv9 = 1.0f / 9.0f;
    #pragma unroll
    for (int v = 0; v < 8; ++v) {
        int o0 = m0 + v + mo;
        int o1 = o0 + 16;
        out[((size_t)bidx * COUT + o0) * HWSZ + p] = acc0[v] * inv9 + bias[o0];
        out[((size_t)bidx * COUT + o1) * HWSZ + p] = acc1[v] * inv9 + bias[o1];
    }
}

// ---------------------------------------------------------------------------
extern "C" void kernel_launch(void* const* d_in, const int* in_sizes, int n_in,
                              void* d_out, int out_size, void* d_ws, size_t ws_size,
                              hipStream_t stream)
{
    const float* x    = (const float*)d_in[0];   // (16,256,56,56)
    const float* G    = (const float*)d_in[1];   // (9,32,256)
    const float* Hm   = (const float*)d_in[2];   // (9,256,32)
    const float* bias = (const float*)d_in[3];   // (256,)
    float* out = (float*)d_out;                  // (16,256,56,56)
    float* U   = (float*)d_ws;                   // 288 * 50176 * 4 = 57.8 MB

    dim3 blk(128);
    lr_stage1<<<dim3(KTOT / 32, NTOT / 64), blk, 0, stream>>>(x, G, U);
    lr_stage2<<<dim3(COUT / 32, NTOT / 64), blk, 0, stream>>>(U, Hm, bias, out);
}